// GraphClassifierNetwork_90503550861772
// MI455X (gfx1250) — compile-verified
//
#include <hip/hip_runtime.h>
#include <hip/hip_bf16.h>

// ---------------------------------------------------------------------------
// GraphClassifierNetwork on gfx1250 (MI455X).
// fp32 end-to-end using V_WMMA_F32_16X16X4_F32 for convs + MLP GEMMs.
// Each wave computes a 16x64 output strip: one A fragment feeds 4 WMMAs,
// cutting VMEM per wmma from 4 to 2.5 and giving the matrix pipe 4
// independent ops per operand fetch (all operands L2-resident).
// ---------------------------------------------------------------------------

typedef float v2f __attribute__((ext_vector_type(2)));
typedef float v8f __attribute__((ext_vector_type(8)));

static __device__ __forceinline__ v8f wmma_f32(v2f a, v2f b, v8f c) {
  // (neg_a, A, neg_b, B, c_mod, C, reuse_a, reuse_b)
  return __builtin_amdgcn_wmma_f32_16x16x4_f32(false, a, false, b,
                                               (short)0, c, false, false);
}

#define NGRAPH 1024
#define SNODES 64
#define NTOT   (NGRAPH * SNODES)   // 65536
#define EPG    1024                // S * DEG edges per graph

// ---------------------------------------------------------------------------
// 1) EdgeConv with the w2-pushdown: segment_sum(h@w2+b2) = segment_sum(h)@w2
//    + deg*b2, so the scatter state is 5 floats/node, held in LDS.
//    One workgroup per graph (edges are block-diagonal).
// ---------------------------------------------------------------------------
__global__ __launch_bounds__(256) void k_edgeconv(
    const float* __restrict__ x, const int* __restrict__ eidx,
    const float* __restrict__ w1, const float* __restrict__ b1,
    const float* __restrict__ w2, const float* __restrict__ b2,
    float* __restrict__ out0, int E_total)
{
  __shared__ float xs[SNODES];
  __shared__ float acc[SNODES * 4];
  __shared__ float cnt[SNODES];
  const int g = blockIdx.x, tid = threadIdx.x;

  if (tid < SNODES) { xs[tid] = x[g * SNODES + tid]; cnt[tid] = 0.f; }
  acc[tid] = 0.f;                       // 256 == 64*4

  float W1[8], B1[4];
#pragma unroll
  for (int k = 0; k < 8; ++k) W1[k] = w1[k];
#pragma unroll
  for (int k = 0; k < 4; ++k) B1[k] = b1[k];
  __syncthreads();

  const int ebase = g * EPG;
  for (int e = tid; e < EPG; e += 256) {
    const int s = eidx[ebase + e]           - g * SNODES;
    const int d = eidx[E_total + ebase + e] - g * SNODES;
    const float xi = xs[d];
    const float xd = xs[s] - xi;
#pragma unroll
    for (int k = 0; k < 4; ++k) {
      float h = fmaxf(xi * W1[k] + xd * W1[4 + k] + B1[k], 0.f);
      atomicAdd(&acc[d * 4 + k], h);
    }
    atomicAdd(&cnt[d], 1.f);
  }
  __syncthreads();

  // out0[node,f] = acc[node,:] @ w2 + deg * b2
  const int node = tid >> 2;
  const int f0   = (tid & 3) * 16;
  const float a0 = acc[node * 4 + 0], a1 = acc[node * 4 + 1];
  const float a2 = acc[node * 4 + 2], a3 = acc[node * 4 + 3];
  const float dc = cnt[node];
  float* orow = out0 + (size_t)(g * SNODES + node) * SNODES;
#pragma unroll
  for (int f = f0; f < f0 + 16; ++f) {
    orow[f] = a0 * w2[f] + a1 * w2[64 + f] + a2 * w2[128 + f] +
              a3 * w2[192 + f] + dc * b2[f];
  }
}

// ---------------------------------------------------------------------------
// 2) BatchNorm stats: one block per channel (no global atomics -> no init).
//    mode 0: X[j*64 + c]  (BN1 over [N,64], channel = column)
//    mode 1: X[(b*64+c)*64 + l], j = b*64+l (conv BN over batch & length)
//    Emits fused scale/shift: y = scale*x + shift.
// ---------------------------------------------------------------------------
__global__ __launch_bounds__(256) void k_bn_stats(
    const float* __restrict__ X, const float* __restrict__ gamma,
    const float* __restrict__ beta, float* __restrict__ scale,
    float* __restrict__ shift, int mode, int count)
{
  const int c = blockIdx.x, tid = threadIdx.x;
  float s = 0.f, ss = 0.f;
  for (int j = tid; j < count; j += 256) {
    float v;
    if (mode == 0) {
      v = X[(size_t)j * 64 + c];
    } else {
      const int b = j >> 6, l = j & 63;
      v = X[((size_t)(b * 64 + c)) * 64 + l];
    }
    s += v; ss += v * v;
  }
  __shared__ float rs[256], rq[256];
  rs[tid] = s; rq[tid] = ss;
  __syncthreads();
  for (int o = 128; o > 0; o >>= 1) {
    if (tid < o) { rs[tid] += rs[tid + o]; rq[tid] += rq[tid + o]; }
    __syncthreads();
  }
  if (tid == 0) {
    const float inv_n = 1.f / (float)count;
    const float mean = rs[0] * inv_n;
    const float var  = rq[0] * inv_n - mean * mean;
    const float sc   = gamma[c] * rsqrtf(var + 1e-5f);
    scale[c] = sc;
    shift[c] = beta[c] - mean * sc;
  }
}

// 3) BN1 apply + ReLU + transpose [N,64] -> channels-first [B,64feat,64node]
__global__ __launch_bounds__(256) void k_bn1_apply_tr(
    const float* __restrict__ X, const float* __restrict__ scale,
    const float* __restrict__ shift, float* __restrict__ Y)
{
  const int g = blockIdx.x;
  for (int j = threadIdx.x; j < 4096; j += 256) {
    const int f = j >> 6, node = j & 63;
    const float v = X[((size_t)(g * 64 + node)) * 64 + f];
    Y[((size_t)(g * 64 + f)) * 64 + node] = fmaxf(scale[f] * v + shift[f], 0.f);
  }
}

// BN apply + ReLU in place on [B,64,64] channels-first
__global__ __launch_bounds__(256) void k_bn_apply(
    float* __restrict__ X, const float* __restrict__ scale,
    const float* __restrict__ shift, int total)
{
  const int i = blockIdx.x * 256 + threadIdx.x;
  if (i < total) {
    const int c = (i >> 6) & 63;
    X[i] = fmaxf(scale[c] * X[i] + shift[c], 0.f);
  }
}

// ---------------------------------------------------------------------------
// 4) Conv1d(64,64,k=3,pad=1) as WMMA GEMM: out[b] = W'[64 x 192] @ X'[192 x 64]
//    K index = t*64 + i ; X'[K,l] = in[b,i,l+t-1] (0 at halo).
//    One block per graph; each wave computes a 64chan x 16pos strip, so the
//    predicated im2col B fragment is built once and reused by 4 WMMAs.
//    A fragments (conv weights) are identical across graphs -> L2/L0 hot.
// ---------------------------------------------------------------------------
__global__ __launch_bounds__(128) void k_conv_wmma(
    const float* __restrict__ in, const float* __restrict__ w,
    const float* __restrict__ bias, float* __restrict__ out)
{
  const int wid  = threadIdx.x >> 5;     // strip: output positions wid*16..+15
  const int lane = threadIdx.x & 31;
  const int b    = blockIdx.x;           // graph
  const int tn   = wid * 16;
  const int half = lane >> 4, lm = lane & 15;
  const int bcol = tn + lm;              // output position this lane feeds B

  const float* inb = in + (size_t)b * 64 * 64;
  v8f c0 = {}, c1 = {}, c2 = {}, c3 = {};
  for (int kk = 0; kk < 192; kk += 4) {
    const int ka = kk + half * 2;
    v2f bb, a0, a1, a2, a3;
#pragma unroll
    for (int q = 0; q < 2; ++q) {
      const int K  = ka + q;
      const int tt = K >> 6;              // tap 0..2
      const int ii = K & 63;              // input channel
      const int lp = bcol + tt - 1;
      bb[q] = (lp >= 0 && lp < 64) ? inb[ii * 64 + lp] : 0.f;
      const float* wq = w + ii * 3 + tt;  // + orow*192
      a0[q] = wq[(lm)      * 192];
      a1[q] = wq[(lm + 16) * 192];
      a2[q] = wq[(lm + 32) * 192];
      a3[q] = wq[(lm + 48) * 192];
    }
    c0 = wmma_f32(a0, bb, c0);
    c1 = wmma_f32(a1, bb, c1);
    c2 = wmma_f32(a2, bb, c2);
    c3 = wmma_f32(a3, bb, c3);
  }
  float* ob = out + (size_t)b * 64 * 64;
#define CONV_STORE(cc, s)                                                  \
  {                                                                        \
    _Pragma("unroll") for (int v = 0; v < 8; ++v) {                        \
      const int o = (s)*16 + v + half * 8;                                 \
      ob[o * 64 + bcol] = (cc)[v] + bias[o];                               \
    }                                                                      \
  }
  CONV_STORE(c0, 0)
  CONV_STORE(c1, 1)
  CONV_STORE(c2, 2)
  CONV_STORE(c3, 3)
#undef CONV_STORE
}

// ---------------------------------------------------------------------------
// 5) WMMA fp32 GEMM, 16x64 strip per wave: C[M,N] = act(A[M,K]@W[K,N]+bias)
//    Requires N % 64 == 0. One A fragment feeds 4 WMMAs per K-step.
// ---------------------------------------------------------------------------
__global__ __launch_bounds__(128) void k_gemm_wmma(
    const float* __restrict__ A, const float* __restrict__ W,
    const float* __restrict__ bias, float* __restrict__ C,
    int M, int N, int K, int relu)
{
  const int wid  = threadIdx.x >> 5;
  const int lane = threadIdx.x & 31;
  const int strip = blockIdx.x * 4 + wid;
  const int nstr  = N >> 6;                 // strips along N
  const int tm    = (strip / nstr) * 16;
  const int tn    = (strip % nstr) * 64;
  const int half  = lane >> 4, lm = lane & 15;

  const float* ap = A + (size_t)(tm + lm) * K;
  v8f c0 = {}, c1 = {}, c2 = {}, c3 = {};
  for (int kk = 0; kk < K; kk += 4) {
    const int ka = kk + half * 2;
    v2f a;
    a.x = ap[ka];
    a.y = ap[ka + 1];
    const float* wp0 = W + (size_t)ka * N + tn + lm;
    const float* wp1 = wp0 + N;
    v2f b0, b1, b2, b3;
    b0.x = wp0[0];  b0.y = wp1[0];
    b1.x = wp0[16]; b1.y = wp1[16];
    b2.x = wp0[32]; b2.y = wp1[32];
    b3.x = wp0[48]; b3.y = wp1[48];
    c0 = wmma_f32(a, b0, c0);
    c1 = wmma_f32(a, b1, c1);
    c2 = wmma_f32(a, b2, c2);
    c3 = wmma_f32(a, b3, c3);
  }
#define GEMM_STORE(cc, s)                                                  \
  {                                                                        \
    const int col = tn + (s)*16 + lm;                                      \
    const float bc = bias[col];                                            \
    _Pragma("unroll") for (int v = 0; v < 8; ++v) {                        \
      const int r = tm + v + half * 8;                                     \
      float val = (cc)[v] + bc;                                            \
      if (relu) val = fmaxf(val, 0.f);                                     \
      C[(size_t)r * N + col] = val;                                        \
    }                                                                      \
  }
  GEMM_STORE(c0, 0)
  GEMM_STORE(c1, 1)
  GEMM_STORE(c2, 2)
  GEMM_STORE(c3, 3)
#undef GEMM_STORE
}

// ---------------------------------------------------------------------------
// 6) Head: [1024,256] @ [256,16] + bias, then row log_softmax.
//    C/D layout puts a full 16-wide row in one VGPR slot across a 16-lane
//    half, so width-16 shfl_xor reductions do the softmax in-register.
// ---------------------------------------------------------------------------
__global__ __launch_bounds__(32) void k_head(
    const float* __restrict__ A, const float* __restrict__ W,
    const float* __restrict__ bias, float* __restrict__ out)
{
  const int lane = threadIdx.x & 31;
  const int tm   = blockIdx.x * 16;
  const int half = lane >> 4, lm = lane & 15;
  const float* ap = A + (size_t)(tm + lm) * 256;

  v8f c = {};
  for (int kk = 0; kk < 256; kk += 4) {
    const int ka = kk + half * 2;
    v2f a, bb;
    a.x  = ap[ka];
    a.y  = ap[ka + 1];
    bb.x = W[ka * 16 + lm];
    bb.y = W[(ka + 1) * 16 + lm];
    c = wmma_f32(a, bb, c);
  }
  const float bc = bias[lm];
#pragma unroll
  for (int v = 0; v < 8; ++v) {
    const float val = c[v] + bc;
    float mx = val;
#pragma unroll
    for (int m = 1; m < 16; m <<= 1) mx = fmaxf(mx, __shfl_xor(mx, m, 16));
    float se = expf(val - mx);
#pragma unroll
    for (int m = 1; m < 16; m <<= 1) se += __shfl_xor(se, m, 16);
    const float res = (val - mx) - logf(se);
    const int r = tm + v + half * 8;
    out[r * 16 + lm] = res;
  }
}

// ---------------------------------------------------------------------------
extern "C" void kernel_launch(void* const* d_in, const int* in_sizes, int n_in,
                              void* d_out, int out_size, void* d_ws, size_t ws_size,
                              hipStream_t stream) {
  const float* x    = (const float*)d_in[0];
  const int*   eidx = (const int*)  d_in[1];
  const float* w1   = (const float*)d_in[2];
  const float* b1   = (const float*)d_in[3];
  const float* w2   = (const float*)d_in[4];
  const float* b2   = (const float*)d_in[5];
  const float* c2w  = (const float*)d_in[6];
  const float* c2b  = (const float*)d_in[7];
  const float* c3w  = (const float*)d_in[8];
  const float* c3b  = (const float*)d_in[9];
  const float* g1   = (const float*)d_in[10];
  const float* be1  = (const float*)d_in[11];
  const float* g2   = (const float*)d_in[12];
  const float* be2  = (const float*)d_in[13];
  const float* g3   = (const float*)d_in[14];
  const float* be3  = (const float*)d_in[15];
  const float* wm1  = (const float*)d_in[16];
  const float* bm1  = (const float*)d_in[17];
  const float* wm2  = (const float*)d_in[18];
  const float* bm2  = (const float*)d_in[19];
  const float* wm3  = (const float*)d_in[20];
  const float* bm3  = (const float*)d_in[21];
  const float* wp   = (const float*)d_in[22];
  const float* bp   = (const float*)d_in[23];

  const int E_total = in_sizes[1] / 2;
  const int FEAT_ELEMS = NGRAPH * 64 * 64;   // 4,194,304 floats (16 MB)

  // Workspace layout: [stats 4KB][bufA 16MB][bufB 16MB]; MLP acts reuse bufA.
  char* ws = (char*)d_ws;
  float* stats = (float*)ws;
  float* bufA  = (float*)(ws + 4096);
  float* bufB  = bufA + (size_t)FEAT_ELEMS;
  float* sc1 = stats,        *sh1 = stats + 64;
  float* sc2 = stats + 128,  *sh2 = stats + 192;
  float* sc3 = stats + 256,  *sh3 = stats + 320;
  float* m1 = bufA;                  // 1024*256
  float* m2 = bufA + 262144;
  float* m3 = bufA + 524288;

  // 1) EdgeConv -> pre-BN node features [N,64] in bufA
  k_edgeconv<<<NGRAPH, 256, 0, stream>>>(x, eidx, w1, b1, w2, b2, bufA, E_total);
  // 2) BN1 (over nodes, per feature column)
  k_bn_stats<<<64, 256, 0, stream>>>(bufA, g1, be1, sc1, sh1, 0, NTOT);
  // 3) apply + ReLU + transpose -> [B,64feat,64node] in bufB
  k_bn1_apply_tr<<<NGRAPH, 256, 0, stream>>>(bufA, sc1, sh1, bufB);
  // 4) conv2 (WMMA) -> bufA   (one block per graph, 4 waves)
  k_conv_wmma<<<NGRAPH, 128, 0, stream>>>(bufB, c2w, c2b, bufA);
  // 5) BN2 + ReLU in place
  k_bn_stats<<<64, 256, 0, stream>>>(bufA, g2, be2, sc2, sh2, 1, NGRAPH * 64);
  k_bn_apply<<<FEAT_ELEMS / 256, 256, 0, stream>>>(bufA, sc2, sh2, FEAT_ELEMS);
  // 6) conv3 (WMMA) -> bufB
  k_conv_wmma<<<NGRAPH, 128, 0, stream>>>(bufA, c3w, c3b, bufB);
  // 7) BN3 + ReLU in place; bufB flattened == [B,4096] feat-major
  k_bn_stats<<<64, 256, 0, stream>>>(bufB, g3, be3, sc3, sh3, 1, NGRAPH * 64);
  k_bn_apply<<<FEAT_ELEMS / 256, 256, 0, stream>>>(bufB, sc3, sh3, FEAT_ELEMS);
  // 8) MLP tail: strips = (M/16)*(N/64); 4 strips per block of 128
  //    MLP1: 64*4 = 256 strips -> 64 blocks; MLP2/3 same.
  k_gemm_wmma<<<64, 128, 0, stream>>>(bufB, wm1, bm1, m1, 1024, 256, 4096, 1);
  k_gemm_wmma<<<64, 128, 0, stream>>>(m1,   wm2, bm2, m2, 1024, 256, 256, 1);
  k_gemm_wmma<<<64, 128, 0, stream>>>(m2,   wm3, bm3, m3, 1024, 256, 256, 1);
  // 9) head GEMM + log_softmax -> d_out [1024,16]
  k_head<<<64, 32, 0, stream>>>(m3, wp, bp, (float*)d_out);
}